// MarkovFlow_26293789786872
// MI455X (gfx1250) — compile-verified
//
#include <hip/hip_runtime.h>
#include <math.h>

// Problem shape (fixed by reference)
#define TT 160
#define BB 512
#define DD 256
#define SS 64

// Workspace layout (float offsets)
#define EXPA_OFF 0                    // SS*SS            = 4096
#define MWT_OFF  4096                 // DD*SS            = 16384
#define MSQ_OFF  20480                // SS               = 64
#define C_OFF    20544                // 1
#define PART_OFF 20608                // BB               = 512
#define IV_OFF   21120                // DD               = 256
#define DENS_OFF 21504                // TT*BB*SS         = 5242880  (~21 MB)

#define EA_STR 72                     // padded row stride for expA in LDS (conflict-free B-frag)
#define XS_STR 260                    // padded row stride for X / ealpha tiles (conflict-free A-frag)

typedef float v2f __attribute__((ext_vector_type(2)));
typedef float v8f __attribute__((ext_vector_type(8)));

// ---------------------------------------------------------------------------
// Prep: inv_var, expA (softmax rows of tparams), mwT = (means*iv)^T, msq, c
// ---------------------------------------------------------------------------
__global__ void mf_prep(const float* __restrict__ means,
                        const float* __restrict__ tparams,
                        const float* __restrict__ var,
                        float* __restrict__ ws) {
  __shared__ float red[256];
  const int tid = threadIdx.x;

  // inv_var + sum(log var)
  float v = (tid < DD) ? var[tid] : 1.0f;
  if (tid < DD) ws[IV_OFF + tid] = 1.0f / v;
  red[tid] = (tid < DD) ? logf(v) : 0.0f;
  __syncthreads();
  for (int s = 128; s > 0; s >>= 1) {
    if (tid < s) red[tid] += red[tid + s];
    __syncthreads();
  }
  if (tid == 0) {
    const float LOG2PI = 1.8378770664093453f;  // log(2) + log(pi)
    ws[C_OFF] = -0.5f * (float)DD * LOG2PI - 0.5f * red[0];
  }

  // mwT[d*SS + s] = means[s][d] * iv[d]
  for (int i = tid; i < DD * SS; i += 256) {
    int d = i / SS, s = i % SS;
    ws[MWT_OFF + i] = means[s * DD + d] * ws[IV_OFF + d];
  }

  if (tid < SS) {
    // msq[s] = sum_d means^2 * iv
    float acc = 0.0f;
    for (int d = 0; d < DD; d++) {
      float m = means[tid * DD + d];
      acc += m * m * ws[IV_OFF + d];
    }
    ws[MSQ_OFF + tid] = acc;

    // expA row = softmax of tparams row
    const float* row = tparams + tid * SS;
    float mx = row[0];
    for (int j = 1; j < SS; j++) mx = fmaxf(mx, row[j]);
    float sum = 0.0f;
    for (int j = 0; j < SS; j++) sum += expf(row[j] - mx);
    float lse = mx + logf(sum);
    for (int j = 0; j < SS; j++) ws[EXPA_OFF + tid * SS + j] = expf(row[j] - lse);
  }
}

// ---------------------------------------------------------------------------
// Pass 1: dens[r][s] = c + x_r . mwT_s - 0.5*xsq[r] - 0.5*msq[s]
// Grid: (TT*BB)/16 blocks of 128 threads (4 waves; wave w -> state cols w*16..)
// X tile staged with CDNA5 async global->LDS DMA (ASYNCcnt).
// ---------------------------------------------------------------------------
__global__ void __launch_bounds__(128)
mf_density(const float* __restrict__ sents,
           const float* __restrict__ ws,
           float* __restrict__ dens) {
  __shared__ float xs[16 * XS_STR];
  __shared__ float xsq[16];
  const int tid = threadIdx.x;
  const long R0 = (long)blockIdx.x * 16;
  const float* xbase = sents + R0 * DD;

  // Stage 16x256 X tile into LDS via async B128 copies (bypasses VGPRs).
  {
    unsigned lds_base = (unsigned)(size_t)&xs[0];   // low 32 bits == LDS offset
    for (int i = tid; i < 16 * (DD / 4); i += 128) {
      int r = i >> 6;                               // DD/4 == 64 chunks per row
      int c4 = i & 63;
      unsigned lds_addr = lds_base + (unsigned)((r * XS_STR + c4 * 4) * 4);
      unsigned goff     = (unsigned)((r * DD + c4 * 4) * 4);
      asm volatile("global_load_async_to_lds_b128 %0, %1, %2"
                   :: "v"(lds_addr), "v"(goff), "s"(xbase)
                   : "memory");
    }
    asm volatile("s_wait_asynccnt 0x0" ::: "memory");
  }
  __syncthreads();

  // Per-row ||x||^2_iv: 16 rows x 8 threads, 32 elems each, shuffle-reduce
  {
    int r = tid >> 3, g = tid & 7;
    const float* iv = ws + IV_OFF;
    float s = 0.0f;
    for (int k = g * 32; k < g * 32 + 32; k++) {
      float x = xs[r * XS_STR + k];
      s += x * x * iv[k];
    }
    s += __shfl_xor(s, 1, 32);
    s += __shfl_xor(s, 2, 32);
    s += __shfl_xor(s, 4, 32);
    if (g == 0) xsq[r] = s;
  }

  const int lane = tid & 31;
  const int w    = tid >> 5;
  const int M    = lane & 15;
  const int hk   = (lane < 16) ? 0 : 2;   // A/B K half per ISA layout
  const int hi   = (lane < 16) ? 0 : 1;   // C/D row half
  const int n    = w * 16 + M;            // output state column
  const float* mwT = ws + MWT_OFF;

  v8f acc = {0.f, 0.f, 0.f, 0.f, 0.f, 0.f, 0.f, 0.f};
  for (int k0 = 0; k0 < DD; k0 += 4) {
    v2f a, b;
    a.x = xs[M * XS_STR + k0 + hk];
    a.y = xs[M * XS_STR + k0 + hk + 1];
    b.x = mwT[(k0 + hk)     * SS + n];
    b.y = mwT[(k0 + hk + 1) * SS + n];
    acc = __builtin_amdgcn_wmma_f32_16x16x4_f32(false, a, false, b,
                                                (short)0, acc, false, false);
  }
  __syncthreads();

  const float cst   = ws[C_OFF];
  const float msq_n = ws[MSQ_OFF + n];
  float* out = dens + R0 * SS;
#pragma unroll
  for (int v = 0; v < 8; v++) {
    int r = v + 8 * hi;
    out[r * SS + n] = cst + acc[v] - 0.5f * xsq[r] - 0.5f * msq_n;
  }
}

// ---------------------------------------------------------------------------
// Pass 2: serial scan over t (latency-bound).
// alpha' = mask*(dens + m + log(exp(alpha-m) @ expA)) + (1-mask)*alpha
// Grid: BB/16 blocks of 128 threads; 2 barriers per step, shuffle row-max,
// prefetch of the next timestep's dens tile.
// ---------------------------------------------------------------------------
__global__ void __launch_bounds__(128)
mf_scan(const float* __restrict__ dens,
        const float* __restrict__ masks,
        const float* __restrict__ ws,
        float* __restrict__ part) {
  __shared__ float eA[SS * EA_STR];       // expA, padded rows
  __shared__ float alpha[16 * XS_STR];
  __shared__ float ealpha[16 * XS_STR];
  __shared__ float mrow[16];
  const int tid = threadIdx.x;
  const int b0  = blockIdx.x * 16;

  for (int i = tid; i < SS * SS; i += 128) {
    int r = i >> 6, c = i & 63;
    eA[r * EA_STR + c] = ws[EXPA_OFF + i];
  }
  const float logS = logf((float)SS);
  for (int i = tid; i < 16 * SS; i += 128) {
    int r = i >> 6, s = i & 63;
    alpha[r * XS_STR + s] = dens[(long)(b0 + r) * SS + s] - logS;  // t=0: log_pi + dens0
  }
  __syncthreads();

  const int lane = tid & 31;
  const int w    = tid >> 5;
  const int M    = lane & 15;
  const int hk   = (lane < 16) ? 0 : 2;
  const int hi   = (lane < 16) ? 0 : 1;
  const int n    = w * 16 + M;
  const int rr   = tid >> 3;              // row owned for max/exp (0..15)
  const int gg   = tid & 7;               // 8-thread group within row

  for (int t = 1; t < TT; t++) {
    // Prefetch next timestep's dens tile (128 threads x 1 cacheline = 16KB)
    {
      const long tp = (t < TT - 1) ? (t + 1) : t;
      __builtin_prefetch(dens + (tp * BB + b0) * SS + tid * 32, 0, 3);
    }
    // Fused row-max + exp (shuffle reduce over 8-lane groups)
    {
      float a0[8];
      float mx = -__builtin_huge_valf();
#pragma unroll
      for (int j = 0; j < 8; j++) {
        a0[j] = alpha[rr * XS_STR + gg * 8 + j];
        mx = fmaxf(mx, a0[j]);
      }
      mx = fmaxf(mx, __shfl_xor(mx, 1, 32));
      mx = fmaxf(mx, __shfl_xor(mx, 2, 32));
      mx = fmaxf(mx, __shfl_xor(mx, 4, 32));
      if (gg == 0) mrow[rr] = mx;
#pragma unroll
      for (int j = 0; j < 8; j++)
        ealpha[rr * XS_STR + gg * 8 + j] = expf(a0[j] - mx);
    }
    __syncthreads();

    // 16x16 tile of ealpha @ expA via fp32 WMMA, K=64
    v8f acc = {0.f, 0.f, 0.f, 0.f, 0.f, 0.f, 0.f, 0.f};
    for (int k0 = 0; k0 < SS; k0 += 4) {
      v2f a, b;
      a.x = ealpha[M * XS_STR + k0 + hk];
      a.y = ealpha[M * XS_STR + k0 + hk + 1];
      b.x = eA[(k0 + hk)     * EA_STR + n];
      b.y = eA[(k0 + hk + 1) * EA_STR + n];
      acc = __builtin_amdgcn_wmma_f32_16x16x4_f32(false, a, false, b,
                                                  (short)0, acc, false, false);
    }

    // alpha update with mask blend (each (r,n) owned by exactly one lane)
    const float* dt = dens  + ((long)t * BB + b0) * SS;
    const float* mt = masks + (long)t * BB + b0;
#pragma unroll
    for (int v = 0; v < 8; v++) {
      int r = v + 8 * hi;
      float mk  = mt[r];
      float nw  = dt[r * SS + n] + mrow[r] + logf(acc[v]);
      float old = alpha[r * XS_STR + n];
      alpha[r * XS_STR + n] = mk * nw + (1.0f - mk) * old;
    }
    __syncthreads();
  }

  // Final per-row logsumexp (same 16x8 decomposition)
  {
    float a0[8];
    float mx = -__builtin_huge_valf();
#pragma unroll
    for (int j = 0; j < 8; j++) {
      a0[j] = alpha[rr * XS_STR + gg * 8 + j];
      mx = fmaxf(mx, a0[j]);
    }
    mx = fmaxf(mx, __shfl_xor(mx, 1, 32));
    mx = fmaxf(mx, __shfl_xor(mx, 2, 32));
    mx = fmaxf(mx, __shfl_xor(mx, 4, 32));
    float sum = 0.0f;
#pragma unroll
    for (int j = 0; j < 8; j++) sum += expf(a0[j] - mx);
    sum += __shfl_xor(sum, 1, 32);
    sum += __shfl_xor(sum, 2, 32);
    sum += __shfl_xor(sum, 4, 32);
    if (gg == 0) part[b0 + rr] = mx + logf(sum);
  }
}

// ---------------------------------------------------------------------------
// Final deterministic reduction (fixed order)
// ---------------------------------------------------------------------------
__global__ void mf_reduce(const float* __restrict__ part, float* __restrict__ out) {
  if (threadIdx.x == 0 && blockIdx.x == 0) {
    float s = 0.0f;
    for (int b = 0; b < BB; b++) s += part[b];
    out[0] = -s;
  }
}

// ---------------------------------------------------------------------------
extern "C" void kernel_launch(void* const* d_in, const int* in_sizes, int n_in,
                              void* d_out, int out_size, void* d_ws, size_t ws_size,
                              hipStream_t stream) {
  (void)in_sizes; (void)n_in; (void)out_size; (void)ws_size;
  const float* sents   = (const float*)d_in[0];   // (T,B,D)
  const float* masks   = (const float*)d_in[1];   // (T,B)
  const float* means   = (const float*)d_in[2];   // (S,D)
  const float* tparams = (const float*)d_in[3];   // (S,S)
  const float* var     = (const float*)d_in[4];   // (D,)
  float* ws  = (float*)d_ws;
  float* out = (float*)d_out;

  mf_prep<<<1, 256, 0, stream>>>(means, tparams, var, ws);
  mf_density<<<(TT * BB) / 16, 128, 0, stream>>>(sents, ws, ws + DENS_OFF);
  mf_scan<<<BB / 16, 128, 0, stream>>>(ws + DENS_OFF, masks, ws, ws + PART_OFF);
  mf_reduce<<<1, 1, 0, stream>>>(ws + PART_OFF, out);
}